// GCN_26199300505693
// MI455X (gfx1250) — compile-verified
//
#include <hip/hip_runtime.h>
#include <hip/hip_bf16.h>
#include <cstddef>

// ---------------------------------------------------------------------------
// GAT (3-layer, single head) for MI455X / gfx1250, wave32.
// GEMMs run on the matrix pipe via V_WMMA_F32_16X16X4_F32 (full fp32 — the
// workload is L2/atomic bound, so fp16 conversion would gain nothing and cost
// accuracy). Edge softmax uses u32-keyed atomic max + f32 atomic adds (L2).
// ---------------------------------------------------------------------------

#define N_NODES 50000
#define E_EDGES 800000
#define E_TOT   (E_EDGES + N_NODES)   // self loops appended
#define NEG_SLOPE 0.2f

typedef float v2f __attribute__((ext_vector_type(2)));
typedef float v8f __attribute__((ext_vector_type(8)));

// ---- monotone float <-> sortable u32 key (for atomic segment-max) ----------
__device__ __forceinline__ unsigned f2key(float f) {
    int i = __float_as_int(f);
    return (i >= 0) ? ((unsigned)i | 0x80000000u) : ~(unsigned)i;
}
__device__ __forceinline__ float key2f(unsigned k) {
    int i = (k & 0x80000000u) ? (int)(k & 0x7FFFFFFFu) : (int)~k;
    return __int_as_float(i);
}

// ---- per-layer state init: acc = 0, mkey = key(-inf-ish) = 0, denom = 0 ----
__global__ void __launch_bounds__(256)
gat_init(float* __restrict__ acc, unsigned* __restrict__ mkey,
         float* __restrict__ denom, int n_nodes, int nf) {
    int i = blockIdx.x * blockDim.x + threadIdx.x;
    int tot = n_nodes * nf;
    if (i < tot) acc[i] = 0.0f;
    if (i < n_nodes) { mkey[i] = 0u; denom[i] = 0.0f; }
}

// ---- H[N,64] = X[N,K] @ W[K,64] via V_WMMA_F32_16X16X4_F32 -----------------
// One wave computes a 16(M) x 64(N) stripe: 4 v8f accumulators, K/4 WMMA steps
// per accumulator. Row tiles divide exactly (50000 = 3125 * 16), so EXEC is
// all-ones inside every participating wave (WMMA requirement).
__global__ void __launch_bounds__(256)
gat_gemm_wmma(const float* __restrict__ X, const float* __restrict__ W,
              float* __restrict__ H, int nrows, int K) {
    const int wave = blockIdx.x * (blockDim.x >> 5) + (threadIdx.x >> 5);
    const int lane = threadIdx.x & 31;
    const int rowBase = wave * 16;
    if (rowBase >= nrows) return;          // wave-uniform: EXEC stays full

    const int lo = lane & 15;              // M (A) / N (B,C,D) within tile
    const int hi = lane >> 4;              // K-pair selector

    v8f acc0 = {}, acc1 = {}, acc2 = {}, acc3 = {};

    const float* xr = X + (size_t)(rowBase + lo) * K + 2 * hi;  // 8B aligned
    const float* wc = W + lo;

    for (int k = 0; k < K; k += 4) {
        // A fragment: lane(0-15): A[m][k..k+1], lane(16-31): A[m][k+2..k+3]
        float2 af = *(const float2*)(xr + k);
        v2f a; a.x = af.x; a.y = af.y;

        // B fragments: v0 = B[k+2*hi][n], v1 = B[k+2*hi+1][n]
        const float* w0 = wc + (size_t)(k + 2 * hi) * 64;
        const float* w1 = w0 + 64;
        v2f b0; b0.x = w0[0];  b0.y = w1[0];
        v2f b1; b1.x = w0[16]; b1.y = w1[16];
        v2f b2; b2.x = w0[32]; b2.y = w1[32];
        v2f b3; b3.x = w0[48]; b3.y = w1[48];

        acc0 = __builtin_amdgcn_wmma_f32_16x16x4_f32(false, a, false, b0, (short)0, acc0, false, false);
        acc1 = __builtin_amdgcn_wmma_f32_16x16x4_f32(false, a, false, b1, (short)0, acc1, false, false);
        acc2 = __builtin_amdgcn_wmma_f32_16x16x4_f32(false, a, false, b2, (short)0, acc2, false, false);
        acc3 = __builtin_amdgcn_wmma_f32_16x16x4_f32(false, a, false, b3, (short)0, acc3, false, false);
    }

    // C/D layout: VGPR r -> rows {r (lanes 0-15), r+8 (lanes 16-31)}, col = lo
#pragma unroll
    for (int r = 0; r < 8; ++r) {
        float* o = H + (size_t)(rowBase + r + 8 * hi) * 64 + lo;
        o[0]  = acc0[r];
        o[16] = acc1[r];
        o[32] = acc2[r];
        o[48] = acc3[r];
    }
}

// ---- per-node attention logits: s[i] = h[i,:]·a_src, d[i] = h[i,:]·a_dst ---
__global__ void __launch_bounds__(256)
gat_node_sd(const float* __restrict__ H, const float* __restrict__ asrc,
            const float* __restrict__ adst, float* __restrict__ s,
            float* __restrict__ d, int n) {
    int i = blockIdx.x * blockDim.x + threadIdx.x;
    if (i >= n) return;
    const float* hr = H + (size_t)i * 64;
    float ss = 0.0f, dd = 0.0f;
#pragma unroll
    for (int f = 0; f < 64; ++f) {
        float h = hr[f];
        ss = fmaf(h, asrc[f], ss);
        dd = fmaf(h, adst[f], dd);
    }
    s[i] = ss; d[i] = dd;
}

// ---- edge pass 1: e = leaky_relu(s[src]+d[dst]); segment max over dst ------
__global__ void __launch_bounds__(256)
gat_edge_max(const int* __restrict__ src, const int* __restrict__ dst,
             const float* __restrict__ s, const float* __restrict__ d,
             float* __restrict__ evals, unsigned* __restrict__ mkey,
             int nE, int etot) {
    int e = blockIdx.x * blockDim.x + threadIdx.x;
    if (e >= etot) return;
    int si = (e < nE) ? src[e] : (e - nE);   // self loops appended at the end
    int di = (e < nE) ? dst[e] : (e - nE);
    float v = s[si] + d[di];
    v = (v > 0.0f) ? v : NEG_SLOPE * v;
    evals[e] = v;
    atomicMax(mkey + di, f2key(v));
}

// ---- edge pass 2: w = exp(e - m[dst]); segment sum of w --------------------
__global__ void __launch_bounds__(256)
gat_edge_expsum(const int* __restrict__ dst, const float* __restrict__ evals,
                const unsigned* __restrict__ mkey, float* __restrict__ wvals,
                float* __restrict__ denom, int nE, int etot) {
    int e = blockIdx.x * blockDim.x + threadIdx.x;
    if (e >= etot) return;
    int di = (e < nE) ? dst[e] : (e - nE);
    float w = __expf(evals[e] - key2f(mkey[di]));
    wvals[e] = w;
    atomicAdd(denom + di, w);
}

// ---- edge pass 3: acc[dst,:] += h[src,:] * (w/denom[dst]) ------------------
// One wave32 per edge: each lane handles features {lane, lane+32}.
__global__ void __launch_bounds__(256)
gat_edge_aggregate(const int* __restrict__ src, const int* __restrict__ dst,
                   const float* __restrict__ H, const float* __restrict__ wvals,
                   const float* __restrict__ denom, float* __restrict__ acc,
                   int nE, int etot) {
    int wid  = blockIdx.x * (blockDim.x >> 5) + (threadIdx.x >> 5);
    int lane = threadIdx.x & 31;
    if (wid >= etot) return;
    int si = (wid < nE) ? src[wid] : (wid - nE);
    int di = (wid < nE) ? dst[wid] : (wid - nE);
    float coef = wvals[wid] / (denom[di] + 1e-16f);
    const float* hr = H + (size_t)si * 64;
    float* ar = acc + (size_t)di * 64;
    atomicAdd(ar + lane,      hr[lane]      * coef);
    atomicAdd(ar + lane + 32, hr[lane + 32] * coef);
}

// ---- bias (+ optional ReLU), acc -> out ------------------------------------
__global__ void __launch_bounds__(256)
gat_bias_act(const float* __restrict__ acc, const float* __restrict__ b,
             float* __restrict__ out, int tot, int relu) {
    int i = blockIdx.x * blockDim.x + threadIdx.x;
    if (i >= tot) return;
    float v = acc[i] + b[i & 63];
    if (relu) v = fmaxf(v, 0.0f);
    out[i] = v;
}

// ---------------------------------------------------------------------------
static void run_gat_layer(const float* xin, int K, const float* W,
                          const float* as, const float* ad, const float* bias,
                          float* H, float* acc, float* nextX, int relu,
                          const int* esrc, const int* edst,
                          float* SV, float* DV, unsigned* MKEY, float* DEN,
                          float* EVAL, float* WVAL, hipStream_t stream) {
    const int nodeFeat = N_NODES * 64;
    // init acc / mkey / denom
    gat_init<<<(nodeFeat + 255) / 256, 256, 0, stream>>>(acc, MKEY, DEN, N_NODES, 64);
    // H = X @ W  (WMMA): 3125 row tiles, 8 waves per 256-thread block
    {
        int waves = N_NODES / 16;                 // 3125, exact
        int blocks = (waves + 7) / 8;
        gat_gemm_wmma<<<blocks, 256, 0, stream>>>(xin, W, H, N_NODES, K);
    }
    gat_node_sd<<<(N_NODES + 255) / 256, 256, 0, stream>>>(H, as, ad, SV, DV, N_NODES);
    gat_edge_max<<<(E_TOT + 255) / 256, 256, 0, stream>>>(esrc, edst, SV, DV, EVAL, MKEY, E_EDGES, E_TOT);
    gat_edge_expsum<<<(E_TOT + 255) / 256, 256, 0, stream>>>(edst, EVAL, MKEY, WVAL, DEN, E_EDGES, E_TOT);
    gat_edge_aggregate<<<(E_TOT + 7) / 8, 256, 0, stream>>>(esrc, edst, H, WVAL, DEN, acc, E_EDGES, E_TOT);
    gat_bias_act<<<(nodeFeat + 255) / 256, 256, 0, stream>>>(acc, bias, nextX, relu, relu);
    // note: 4th arg of gat_bias_act is `tot` -- fix below in call sites
}

extern "C" void kernel_launch(void* const* d_in, const int* in_sizes, int n_in,
                              void* d_out, int out_size, void* d_ws, size_t ws_size,
                              hipStream_t stream) {
    (void)in_sizes; (void)n_in; (void)out_size; (void)ws_size;

    const float* x    = (const float*)d_in[0];   // [N,128]
    const int*   esrc = (const int*)  d_in[1];   // [E]
    const int*   edst = (const int*)  d_in[2];   // [E]
    const float* W0 = (const float*)d_in[3];
    const float* a0s = (const float*)d_in[4];
    const float* a0d = (const float*)d_in[5];
    const float* b0 = (const float*)d_in[6];
    const float* W1 = (const float*)d_in[7];
    const float* a1s = (const float*)d_in[8];
    const float* a1d = (const float*)d_in[9];
    const float* b1 = (const float*)d_in[10];
    const float* W2 = (const float*)d_in[11];
    const float* a2s = (const float*)d_in[12];
    const float* a2d = (const float*)d_in[13];
    const float* b2 = (const float*)d_in[14];
    float* out = (float*)d_out;                  // [N,64]

    // workspace layout (all fp32/u32, 4B aligned)
    float* ws = (float*)d_ws;
    float*    H    = ws;                                // N*64
    float*    XIN  = H   + (size_t)N_NODES * 64;        // N*64
    float*    ACC  = XIN + (size_t)N_NODES * 64;        // N*64
    float*    SV   = ACC + (size_t)N_NODES * 64;        // N
    float*    DV   = SV  + N_NODES;                     // N
    unsigned* MKEY = (unsigned*)(DV + N_NODES);         // N
    float*    DEN  = (float*)(MKEY + N_NODES);          // N
    float*    EVAL = DEN + N_NODES;                     // E_TOT
    float*    WVAL = EVAL + E_TOT;                      // E_TOT

    const int nodeFeat = N_NODES * 64;

    // ---- layer 0: IN=128 -> 64, ReLU -> XIN ----
    gat_init<<<(nodeFeat + 255) / 256, 256, 0, stream>>>(ACC, MKEY, DEN, N_NODES, 64);
    gat_gemm_wmma<<<((N_NODES / 16) + 7) / 8, 256, 0, stream>>>(x, W0, H, N_NODES, 128);
    gat_node_sd<<<(N_NODES + 255) / 256, 256, 0, stream>>>(H, a0s, a0d, SV, DV, N_NODES);
    gat_edge_max<<<(E_TOT + 255) / 256, 256, 0, stream>>>(esrc, edst, SV, DV, EVAL, MKEY, E_EDGES, E_TOT);
    gat_edge_expsum<<<(E_TOT + 255) / 256, 256, 0, stream>>>(edst, EVAL, MKEY, WVAL, DEN, E_EDGES, E_TOT);
    gat_edge_aggregate<<<(E_TOT + 7) / 8, 256, 0, stream>>>(esrc, edst, H, WVAL, DEN, ACC, E_EDGES, E_TOT);
    gat_bias_act<<<(nodeFeat + 255) / 256, 256, 0, stream>>>(ACC, b0, XIN, nodeFeat, 1);

    // ---- layer 1: 64 -> 64, ReLU -> XIN (in place over dead input) ----
    gat_init<<<(nodeFeat + 255) / 256, 256, 0, stream>>>(ACC, MKEY, DEN, N_NODES, 64);
    gat_gemm_wmma<<<((N_NODES / 16) + 7) / 8, 256, 0, stream>>>(XIN, W1, H, N_NODES, 64);
    gat_node_sd<<<(N_NODES + 255) / 256, 256, 0, stream>>>(H, a1s, a1d, SV, DV, N_NODES);
    gat_edge_max<<<(E_TOT + 255) / 256, 256, 0, stream>>>(esrc, edst, SV, DV, EVAL, MKEY, E_EDGES, E_TOT);
    gat_edge_expsum<<<(E_TOT + 255) / 256, 256, 0, stream>>>(edst, EVAL, MKEY, WVAL, DEN, E_EDGES, E_TOT);
    gat_edge_aggregate<<<(E_TOT + 7) / 8, 256, 0, stream>>>(esrc, edst, H, WVAL, DEN, ACC, E_EDGES, E_TOT);
    gat_bias_act<<<(nodeFeat + 255) / 256, 256, 0, stream>>>(ACC, b1, XIN, nodeFeat, 1);

    // ---- layer 2: 64 -> 64, no ReLU, result -> d_out ----
    gat_init<<<(nodeFeat + 255) / 256, 256, 0, stream>>>(out, MKEY, DEN, N_NODES, 64);
    gat_gemm_wmma<<<((N_NODES / 16) + 7) / 8, 256, 0, stream>>>(XIN, W2, H, N_NODES, 64);
    gat_node_sd<<<(N_NODES + 255) / 256, 256, 0, stream>>>(H, a2s, a2d, SV, DV, N_NODES);
    gat_edge_max<<<(E_TOT + 255) / 256, 256, 0, stream>>>(esrc, edst, SV, DV, EVAL, MKEY, E_EDGES, E_TOT);
    gat_edge_expsum<<<(E_TOT + 255) / 256, 256, 0, stream>>>(edst, EVAL, MKEY, WVAL, DEN, E_EDGES, E_TOT);
    gat_edge_aggregate<<<(E_TOT + 7) / 8, 256, 0, stream>>>(esrc, edst, H, WVAL, DEN, out, E_EDGES, E_TOT);
    gat_bias_act<<<(nodeFeat + 255) / 256, 256, 0, stream>>>(out, b2, out, nodeFeat, 0);
}